// FastDnXSurrogateModel_45483703665112
// MI455X (gfx1250) — compile-verified
//
#include <hip/hip_runtime.h>
#include <stdint.h>

#define N_NODES  50000
#define N_EDGES  800000
#define IN_FEAT  128
#define N_CLASSES 64

typedef __attribute__((ext_vector_type(2))) float v2f;
typedef __attribute__((ext_vector_type(8))) float v8f;

// ---------------- degree / normalization ----------------

__global__ void deg_init_kernel(float* __restrict__ deg) {
  int i = blockIdx.x * blockDim.x + threadIdx.x;
  if (i < N_NODES) deg[i] = 1.0f;  // self-loop contribution
}

__global__ void deg_count_kernel(const long long* __restrict__ col,
                                 float* __restrict__ deg) {
  int e = blockIdx.x * blockDim.x + threadIdx.x;
  if (e < N_EDGES) atomicAdd(&deg[(int)col[e]], 1.0f);
}

__global__ void dinv_kernel(float* __restrict__ deg) {
  int i = blockIdx.x * blockDim.x + threadIdx.x;
  if (i < N_NODES) deg[i] = rsqrtf(deg[i]);  // deg >= 1 always
}

// ---------------- propagation (one hop) ----------------

// h_out[i] = dinv[i]^2 * h_in[i]   (self-loop term; also initializes h_out)
__global__ void selfloop_kernel(const float* __restrict__ h_in,
                                float* __restrict__ h_out,
                                const float* __restrict__ dinv) {
  int idx = blockIdx.x * blockDim.x + threadIdx.x;  // over N_NODES * 32 float4s
  if (idx >= N_NODES * (IN_FEAT / 4)) return;
  int node = idx >> 5;                 // 32 float4 per row
  float d = dinv[node];
  float s = d * d;
  float4 v = ((const float4*)h_in)[idx];
  v.x *= s; v.y *= s; v.z *= s; v.w *= s;
  ((float4*)h_out)[idx] = v;
}

// one wave per edge; lane handles 4 consecutive features
__global__ void spmm_kernel(const long long* __restrict__ row,
                            const long long* __restrict__ col,
                            const float* __restrict__ dinv,
                            const float* __restrict__ h_in,
                            float* __restrict__ h_out) {
  int e    = blockIdx.x * (blockDim.x >> 5) + (threadIdx.x >> 5);  // wave-uniform
  int lane = threadIdx.x & 31;
  if (e >= N_EDGES) return;
  int r = (int)row[e];
  int c = (int)col[e];
  float norm = dinv[r] * dinv[c];
  float4 v = ((const float4*)(h_in + (size_t)r * IN_FEAT))[lane];
  float* dst = h_out + (size_t)c * IN_FEAT + lane * 4;
  atomicAdd(dst + 0, norm * v.x);
  atomicAdd(dst + 1, norm * v.y);
  atomicAdd(dst + 2, norm * v.z);
  atomicAdd(dst + 3, norm * v.w);
}

// ---------------- WMMA GEMM: out = h @ W^T + b ----------------
// Tile: 16 (nodes) x 16 (classes) per wave; K=128 in steps of 4 using
// V_WMMA_F32_16X16X4_F32 (fp32 in, fp32 accumulate).
// A (16x4) layout: lane<16 -> row=lane, K={0,1}; lane>=16 -> row=lane-16, K={2,3}
// B (4x16)  layout: lane<16 -> col=lane, K={0,1}; lane>=16 -> col=lane-16, K={2,3}
// C/D layout: VGPR i -> row = i + 8*(lane>>4), col = lane&15
__global__ void gemm_wmma_kernel(const float* __restrict__ h,
                                 const float* __restrict__ W,
                                 const float* __restrict__ b,
                                 float* __restrict__ out) {
  const int lane = threadIdx.x & 31;
  const int wave = threadIdx.x >> 5;
  const int tile = blockIdx.x * 4 + wave;      // 12500 tiles total, exact
  const int nt = tile & 3;                     // 64 classes / 16
  const int mt = tile >> 2;                    // 50000 nodes / 16 = 3125
  const int m0 = mt * 16;
  const int n0 = nt * 16;
  const int half = lane >> 4;                  // 0 or 1
  const int l15  = lane & 15;
  const int koff = half * 2;                   // K pair: {0,1} or {2,3}
  const int m = m0 + l15;
  const int n = n0 + l15;

  const float* __restrict__ arow = h + (size_t)m * IN_FEAT + koff;
  const float* __restrict__ brow = W + (size_t)n * IN_FEAT + koff;  // B[k][n] = W[n][k]

  v8f acc = {};
#pragma unroll
  for (int k0 = 0; k0 < IN_FEAT; k0 += 4) {
    v2f a  = *(const v2f*)(arow + k0);
    v2f bb = *(const v2f*)(brow + k0);
    acc = __builtin_amdgcn_wmma_f32_16x16x4_f32(
        /*neg_a=*/false, a, /*neg_b=*/false, bb,
        /*c_mod=*/(short)0, acc, /*reuse_a=*/false, /*reuse_b=*/false);
  }

  float bias = b[n];
#pragma unroll
  for (int i = 0; i < 8; ++i) {
    int r = m0 + i + half * 8;
    out[(size_t)r * N_CLASSES + n] = acc[i] + bias;
  }
}

// ---------------- launch ----------------

extern "C" void kernel_launch(void* const* d_in, const int* in_sizes, int n_in,
                              void* d_out, int out_size, void* d_ws, size_t ws_size,
                              hipStream_t stream) {
  const float*     x  = (const float*)d_in[0];
  const long long* ei = (const long long*)d_in[1];  // int64 [2, N_EDGES]
  const float*     W  = (const float*)d_in[2];      // [64, 128]
  const float*     b  = (const float*)d_in[3];      // [64]
  float* out = (float*)d_out;

  const long long* row = ei;             // edge_index[0] : sources
  const long long* col = ei + N_EDGES;   // edge_index[1] : targets

  // workspace layout (floats): dinv | h1 | h2
  float* dinv = (float*)d_ws;
  float* h1 = dinv + ((N_NODES + 63) & ~63);
  float* h2 = h1 + (size_t)N_NODES * IN_FEAT;

  // normalization
  deg_init_kernel<<<(N_NODES + 255) / 256, 256, 0, stream>>>(dinv);
  deg_count_kernel<<<(N_EDGES + 255) / 256, 256, 0, stream>>>(col, dinv);
  dinv_kernel<<<(N_NODES + 255) / 256, 256, 0, stream>>>(dinv);

  const int selfloop_threads = N_NODES * (IN_FEAT / 4);
  const int spmm_blocks = (N_EDGES * 32 + 255) / 256;

  // hop 1: x -> h1
  selfloop_kernel<<<(selfloop_threads + 255) / 256, 256, 0, stream>>>(x, h1, dinv);
  spmm_kernel<<<spmm_blocks, 256, 0, stream>>>(row, col, dinv, x, h1);

  // hop 2: h1 -> h2
  selfloop_kernel<<<(selfloop_threads + 255) / 256, 256, 0, stream>>>(h1, h2, dinv);
  spmm_kernel<<<spmm_blocks, 256, 0, stream>>>(row, col, dinv, h1, h2);

  // dense tail: out = h2 @ W^T + b   (12500 tiles, 4 waves/block -> 3125 blocks)
  gemm_wmma_kernel<<<3125, 128, 0, stream>>>(h2, W, b, out);
}